// aspect_augumentation_book_18511309046227
// MI455X (gfx1250) — compile-verified
//
#include <hip/hip_runtime.h>
#include <hip/hip_bf16.h>

typedef float v2f __attribute__((ext_vector_type(2)));
typedef float v8f __attribute__((ext_vector_type(8)));

#define D 64
#define N_RK 3
#define LA 20
#define LC 5

// ---------------------------------------------------------------------------
// Kernel A: scores = softmax(leaky_relu(users @ relation_k), axis=1)
// One wave (32 threads) per 16 batch rows. Full-precision f32 WMMA:
// 16 chained V_WMMA_F32_16X16X4_F32 over K=64.
//
// A tile (16x4 f32, 2 VGPRs):  lane m=lane&15, half=lane>>4
//   v0 = U[m][k0 + 2*half], v1 = U[m][k0 + 2*half + 1]
// B tile (4x16 f32, 2 VGPRs):  col n=lane&15
//   v0 = R[k0 + 2*half][n], v1 = R[k0 + 2*half + 1][n]   (n>=3 -> 0)
// C/D (16x16 f32, 8 VGPRs):    d[r] = P[r + 8*half][n]
//
// B-operand columns >=3 are zeroed with a branch-free clamped-index load
// times a 0/1 lane mask (no exec-mask divergence between WMMAs).
// ---------------------------------------------------------------------------
__global__ __launch_bounds__(32) void scores_wmma_kernel(
    const int* __restrict__ user_id,
    const float* __restrict__ user_factors,
    const float* __restrict__ relation_k,
    float* __restrict__ scores_out,   // (B,3)
    int B)
{
    __shared__ float tile[16][17];    // +1 pad: no bank conflicts

    const int lane = threadIdx.x & 31;
    const int m    = lane & 15;
    const int half = lane >> 4;       // 0: K pair {0,1}; 1: K pair {2,3}
    const int n    = m;               // output column this lane owns
    const int b0   = blockIdx.x * 16;

    // gather this lane's user row (lanes m and m+16 share row m)
    int bi  = b0 + m; if (bi >= B) bi = B - 1;
    const float* __restrict__ urow =
        user_factors + (size_t)user_id[bi] * D;

    // branch-free B-operand addressing: clamp column, mask value
    const float nmask = (n < N_RK) ? 1.0f : 0.0f;
    const int   nc    = (n < N_RK) ? n : 0;

    v8f acc = {};
    #pragma unroll
    for (int k0 = 0; k0 < D; k0 += 4) {
        const int kb = k0 + half * 2;
        v2f a;
        a.x = urow[kb];
        a.y = urow[kb + 1];
        v2f bm;
        bm.x = relation_k[kb * N_RK + nc]       * nmask;
        bm.y = relation_k[(kb + 1) * N_RK + nc] * nmask;
        // D = A x B + C, f32 in / f32 accumulate (exact vs reference)
        acc = __builtin_amdgcn_wmma_f32_16x16x4_f32(
            /*neg_a=*/false, a, /*neg_b=*/false, bm,
            /*c_mod=*/(short)0, acc, /*reuse_a=*/false, /*reuse_b=*/false);
    }

    // stage D back: VGPR r holds row (r + 8*half), column n
    #pragma unroll
    for (int r = 0; r < 8; ++r)
        tile[r + half * 8][n] = acc[r];
    __syncthreads();

    // 16 lanes finish one row each: leaky_relu -> stable softmax over 3
    if (lane < 16) {
        const int b = b0 + lane;
        if (b < B) {
            float x0 = tile[lane][0];
            float x1 = tile[lane][1];
            float x2 = tile[lane][2];
            x0 = (x0 > 0.0f) ? x0 : 0.2f * x0;
            x1 = (x1 > 0.0f) ? x1 : 0.2f * x1;
            x2 = (x2 > 0.0f) ? x2 : 0.2f * x2;
            const float mx = fmaxf(x0, fmaxf(x1, x2));
            const float e0 = __expf(x0 - mx);
            const float e1 = __expf(x1 - mx);
            const float e2 = __expf(x2 - mx);
            const float inv = 1.0f / (e0 + e1 + e2);
            scores_out[b * 3 + 0] = e0 * inv;
            scores_out[b * 3 + 1] = e1 * inv;
            scores_out[b * 3 + 2] = e2 * inv;
        }
    }
}

// ---------------------------------------------------------------------------
// Kernel B: gather entity rows, dot with user vector, wave32 reduce.
// One wave per batch item (8 waves / 256-thread block). Each lane holds
// user[d=lane] and user[d=lane+32]; every entity-row read is two coalesced
// 128B transactions (L2-resident table: 25.6 MB << 192 MB L2).
// ---------------------------------------------------------------------------
__global__ __launch_bounds__(256) void contrib_kernel(
    const int*   __restrict__ user_id,
    const int*   __restrict__ artists_id,     // (B, LA)
    const int*   __restrict__ categories_id,  // (B, LC)
    const float* __restrict__ user_factors,
    const float* __restrict__ entity_factors,
    const float* __restrict__ scores,         // (B,3) written by kernel A
    float* __restrict__ prediction,           // (B)
    float* __restrict__ c_act_out,            // (B)
    float* __restrict__ c_dir_out,            // (B)
    float* __restrict__ niubi_act,            // (B, LA)
    float* __restrict__ niubi_dir,            // (B, LC)
    int B)
{
    const int lane = threadIdx.x & 31;
    const int wave = threadIdx.x >> 5;
    const int b    = blockIdx.x * 8 + wave;
    if (b >= B) return;

    const float* __restrict__ urow =
        user_factors + (size_t)user_id[b] * D;
    const float u0 = urow[lane];
    const float u1 = urow[lane + 32];

    float sumA = 0.0f;
    #pragma unroll 4
    for (int l = 0; l < LA; ++l) {
        const int idx = artists_id[b * LA + l];       // wave-uniform broadcast
        const float* __restrict__ e = entity_factors + (size_t)idx * D;
        float p = u0 * e[lane] + u1 * e[lane + 32];
        #pragma unroll
        for (int o = 16; o > 0; o >>= 1)
            p += __shfl_xor(p, o, 32);                // butterfly: all lanes hold sum
        if (lane == 0) niubi_act[b * LA + l] = p;
        sumA += p;
    }

    float sumD = 0.0f;
    #pragma unroll
    for (int l = 0; l < LC; ++l) {
        const int idx = categories_id[b * LC + l];
        const float* __restrict__ e = entity_factors + (size_t)idx * D;
        float p = u0 * e[lane] + u1 * e[lane + 32];
        #pragma unroll
        for (int o = 16; o > 0; o >>= 1)
            p += __shfl_xor(p, o, 32);
        if (lane == 0) niubi_dir[b * LC + l] = p;
        sumD += p;
    }

    if (lane == 0) {
        const float ca = sumA * (1.0f / LA);
        const float cd = sumD * (1.0f / LC);
        const float sa = scores[b * 3 + 0];
        const float sd = scores[b * 3 + 1];
        c_act_out[b]  = ca;
        c_dir_out[b]  = cd;
        prediction[b] = (ca * sa + cd * sd) / (sa + sd);
    }
}

extern "C" void kernel_launch(void* const* d_in, const int* in_sizes, int n_in,
                              void* d_out, int out_size, void* d_ws, size_t ws_size,
                              hipStream_t stream) {
    const int*   user_id        = (const int*)  d_in[0];
    const int*   artists_id     = (const int*)  d_in[1];
    const int*   categories_id  = (const int*)  d_in[2];
    // d_in[3] = rate (unused scalar)
    const float* user_factors   = (const float*)d_in[4];
    const float* entity_factors = (const float*)d_in[5];
    const float* relation_k     = (const float*)d_in[6];

    const int B = in_sizes[0];

    // Output tuple, flat: pred | scores | c_act | c_dir | niubi_act | niubi_dir
    float* out        = (float*)d_out;
    float* pred       = out;
    float* scores     = out + (size_t)B;
    float* c_act      = out + (size_t)4 * B;
    float* c_dir      = out + (size_t)5 * B;
    float* niubi_act  = out + (size_t)6 * B;
    float* niubi_dir  = out + (size_t)26 * B;

    // Kernel A: WMMA f32 scores (one wave per 16 rows)
    const int gridA = (B + 15) / 16;
    scores_wmma_kernel<<<gridA, 32, 0, stream>>>(
        user_id, user_factors, relation_k, scores, B);

    // Kernel B: gather + dot + reduce (one wave per item, 8 waves/block)
    const int gridB = (B + 7) / 8;
    contrib_kernel<<<gridB, 256, 0, stream>>>(
        user_id, artists_id, categories_id, user_factors, entity_factors,
        scores, pred, c_act, c_dir, niubi_act, niubi_dir, B);
}